// DynamicGAT_29025388987029
// MI455X (gfx1250) — compile-verified
//
#include <hip/hip_runtime.h>
#include <hip/hip_bf16.h>

// ---------------------------------------------------------------------------
// DynamicGAT pipeline for MI455X (gfx1250, wave32).
// Dense GEMMs use V_WMMA_F32_16X16X4_F32 (exact fp32 -> matches f32 reference;
// GEMM flops are tiny vs the L2-resident edge scatter/gather, so no precision
// downcast is warranted). Edge softmax/aggregation uses order-preserving uint
// atomicMax for segment-max and f32 global atomics for segment-sum.
// ---------------------------------------------------------------------------

typedef __attribute__((ext_vector_type(2))) float v2f;
typedef __attribute__((ext_vector_type(8))) float v8f;

__device__ __forceinline__ unsigned fenc(float f) {
    unsigned u = __float_as_uint(f);
    return (u & 0x80000000u) ? ~u : (u | 0x80000000u);
}
__device__ __forceinline__ float fdec(unsigned u) {
    return (u & 0x80000000u) ? __uint_as_float(u ^ 0x80000000u)
                             : __uint_as_float(~u);
}
__device__ __forceinline__ float lrelu(float x) { return x > 0.f ? x : 0.2f * x; }

// ---------------- GraphConv -------------------------------------------------
// agg[dst] += x[src]  (IN = 2 floats per node)
__global__ void gc_edge_kernel(const float* __restrict__ x,
                               const int* __restrict__ ei,
                               float* __restrict__ agg, int E) {
    int e = blockIdx.x * blockDim.x + threadIdx.x;
    if (e >= E) return;
    int s = ei[e];
    int d = ei[E + e];
    atomicAdd(&agg[d * 2 + 0], x[s * 2 + 0]);
    atomicAdd(&agg[d * 2 + 1], x[s * 2 + 1]);
}

// h0 = relu(agg @ W_rel + b_rel + x @ W_root), K=2 so scalar math is optimal
__global__ void gc_node_kernel(const float* __restrict__ x,
                               const float* __restrict__ agg,
                               const float* __restrict__ Wrel,
                               const float* __restrict__ brel,
                               const float* __restrict__ Wroot,
                               float* __restrict__ h, int N) {
    int i = blockIdx.x * blockDim.x + threadIdx.x;
    if (i >= N * 64) return;
    int n = i >> 6, c = i & 63;
    float a0 = agg[n * 2], a1 = agg[n * 2 + 1];
    float x0 = x[n * 2],   x1 = x[n * 2 + 1];
    float v = a0 * Wrel[c] + a1 * Wrel[64 + c] + brel[c]
            + x0 * Wroot[c] + x1 * Wroot[64 + c];
    h[i] = fmaxf(v, 0.f);
}

// ---------------- fp32 WMMA GEMM:  H[M,F] = X[M,K] @ W[K,F] -----------------
// One wave per 16x16 output tile. K stepped by 4 per V_WMMA_F32_16X16X4_F32.
// A frag (16x4): lane<16 holds row m=lane, K=k..k+1; lane>=16 row m=lane-16,
//                K=k+2..k+3 -> contiguous float2 load from row-major X.
// B frag (4x16): VGPR0 = rows k / k+2, VGPR1 = rows k+1 / k+3 (strided loads).
__global__ void __launch_bounds__(32)
gemm_wmma_f32(const float* __restrict__ X, const float* __restrict__ W,
              float* __restrict__ H, int K, int F) {
    const int bm   = blockIdx.x * 16;
    const int bn   = blockIdx.y * 16;
    const int lane = threadIdx.x;
    const int half = lane >> 4;   // 0: lanes 0-15, 1: lanes 16-31
    const int l    = lane & 15;

    const float* xrow = X + (size_t)(bm + l) * K + 2 * half;
    const float* wcol = W + bn + l;

    v8f acc = {};
    for (int k = 0; k < K; k += 4) {
        v2f a = *(const v2f*)(xrow + k);                 // K=k+2h, k+2h+1
        v2f b;
        b.x = wcol[(size_t)(k + 2 * half) * F];          // row k or k+2
        b.y = wcol[(size_t)(k + 2 * half + 1) * F];      // row k+1 or k+3
        acc = __builtin_amdgcn_wmma_f32_16x16x4_f32(
            /*neg_a=*/false, a, /*neg_b=*/false, b,
            /*c_mod=*/(short)0, acc, /*reuse_a=*/false, /*reuse_b=*/false);
    }
    // C/D layout: VGPR r -> rows (r, r+8); lanes 0-15 / 16-31 pick the half.
    float* hrow = H + (size_t)(bm + 8 * half) * F + bn + l;
#pragma unroll
    for (int r = 0; r < 8; ++r) hrow[(size_t)r * F] = acc[r];
}

// scalar tail for M rows not covered by 16-row tiles (insurance; N=50000 is
// divisible by 16 so this normally launches nothing)
__global__ void gemm_tail(const float* __restrict__ X, const float* __restrict__ W,
                          float* __restrict__ H, int K, int F, int row0, int N) {
    int i = blockIdx.x * blockDim.x + threadIdx.x;
    int total = (N - row0) * F;
    if (i >= total) return;
    int r = row0 + i / F, c = i % F;
    float acc = 0.f;
    for (int k = 0; k < K; ++k) acc += X[(size_t)r * K + k] * W[(size_t)k * F + c];
    H[(size_t)r * F + c] = acc;
}

// ---------------- GAT attention passes --------------------------------------
// per (node, head): alpha_src/dst dots; init segment-max with the self-loop
// edge value (every node has one), zero the denominator.
__global__ void alphas_kernel(const float* __restrict__ H,
                              const float* __restrict__ a_s,
                              const float* __restrict__ a_d,
                              float* __restrict__ al_s, float* __restrict__ al_d,
                              unsigned* __restrict__ m_enc, float* __restrict__ den,
                              int N, int heads, int oc) {
    int i = blockIdx.x * blockDim.x + threadIdx.x;
    if (i >= N * heads) return;
    int n = i / heads, hh = i - n * heads;
    const float* hr = H + (size_t)n * heads * oc + (size_t)hh * oc;
    const float* as = a_s + hh * oc;
    const float* ad = a_d + hh * oc;
    float s = 0.f, d = 0.f;
    for (int c = 0; c < oc; ++c) { float v = hr[c]; s += v * as[c]; d += v * ad[c]; }
    al_s[i] = s;
    al_d[i] = d;
    m_enc[i] = fenc(lrelu(s + d));   // self-loop edge seeds the max
    den[i]   = 0.f;
}

// segment max over the E real edges (self-loops already folded in above)
__global__ void edge_max_kernel(const int* __restrict__ ei,
                                const float* __restrict__ al_s,
                                const float* __restrict__ al_d,
                                unsigned* __restrict__ m_enc, int E, int heads) {
    int idx = blockIdx.x * blockDim.x + threadIdx.x;
    if (idx >= E * heads) return;
    int e = idx / heads, hh = idx - e * heads;
    int s = ei[e], d = ei[E + e];
    float ev = lrelu(al_s[s * heads + hh] + al_d[d * heads + hh]);
    atomicMax(&m_enc[d * heads + hh], fenc(ev));
}

// w = exp(e - m[dst]); den[dst] += w; stash w per (edge, head)
__global__ void edge_w_kernel(const int* __restrict__ ei,
                              const float* __restrict__ al_s,
                              const float* __restrict__ al_d,
                              const unsigned* __restrict__ m_enc,
                              float* __restrict__ den, float* __restrict__ wbuf,
                              int E, int N, int heads) {
    int idx = blockIdx.x * blockDim.x + threadIdx.x;
    int total = (E + N) * heads;
    if (idx >= total) return;
    int e = idx / heads, hh = idx - e * heads;
    int s, d;
    if (e < E) { s = ei[e]; d = ei[E + e]; } else { s = d = e - E; }
    float ev = lrelu(al_s[s * heads + hh] + al_d[d * heads + hh]);
    float m  = fdec(m_enc[d * heads + hh]);
    float w  = __expf(ev - m);
    wbuf[idx] = w;
    atomicAdd(&den[d * heads + hh], w);
}

// out[dst] += h[src] * alpha  — one wave32 per edge, 128B coalesced gathers of
// h[src] (L2-resident), f32 atomic scatter into dst rows.
__global__ void edge_aggr_kernel(const int* __restrict__ ei,
                                 const float* __restrict__ H,
                                 const float* __restrict__ wbuf,
                                 const float* __restrict__ den,
                                 float* __restrict__ out,
                                 int E, int N, int heads, int oc) {
    const int F    = heads * oc;
    const int wave = threadIdx.x >> 5;        // 8 waves / block
    const int lane = threadIdx.x & 31;
    long long e = (long long)blockIdx.x * 8 + wave;
    if (e >= (long long)E + N) return;
    int s, d;
    if (e < E) { s = ei[(int)e]; d = ei[E + (int)e]; } else { s = d = (int)(e - E); }
    const float* hs = H + (size_t)s * F;
    float* od = out + (size_t)d * F;
    for (int c = lane; c < F; c += 32) {
        int hh = c / oc;
        float coef = wbuf[e * heads + hh] / (den[d * heads + hh] + 1e-16f);
        atomicAdd(&od[c], hs[c] * coef);
    }
}

__global__ void finalize_kernel(float* __restrict__ out, const float* __restrict__ b,
                                long long total, int F, int do_relu) {
    long long i = (long long)blockIdx.x * blockDim.x + threadIdx.x;
    if (i >= total) return;
    float v = out[i] + b[(int)(i % F)];
    out[i] = do_relu ? fmaxf(v, 0.f) : v;
}

// ---------------- host-side layer driver ------------------------------------
static void run_gat_layer(const float* Xin, const float* W, const float* a_s,
                          const float* a_d, const float* b, int N, int E, int K,
                          int heads, int oc, int do_relu, const int* ei,
                          float* H, float* out, float* al_s, float* al_d,
                          unsigned* m_enc, float* den, float* wbuf,
                          hipStream_t stream) {
    const int F = heads * oc;
    const int Mt = (N / 16) * 16;
    if (Mt > 0) {
        dim3 gg(Mt / 16, F / 16);
        gemm_wmma_f32<<<gg, 32, 0, stream>>>(Xin, W, H, K, F);
    }
    if (Mt < N) {
        int tail = (N - Mt) * F;
        gemm_tail<<<(tail + 255) / 256, 256, 0, stream>>>(Xin, W, H, K, F, Mt, N);
    }
    int nh = N * heads;
    alphas_kernel<<<(nh + 255) / 256, 256, 0, stream>>>(H, a_s, a_d, al_s, al_d,
                                                        m_enc, den, N, heads, oc);
    hipMemsetAsync(out, 0, (size_t)N * F * sizeof(float), stream);
    int eh = E * heads;
    edge_max_kernel<<<(eh + 255) / 256, 256, 0, stream>>>(ei, al_s, al_d, m_enc, E, heads);
    int th = (E + N) * heads;
    edge_w_kernel<<<(th + 255) / 256, 256, 0, stream>>>(ei, al_s, al_d, m_enc, den,
                                                        wbuf, E, N, heads);
    long long ne = (long long)E + N;
    edge_aggr_kernel<<<(unsigned)((ne + 7) / 8), 256, 0, stream>>>(ei, H, wbuf, den,
                                                                   out, E, N, heads, oc);
    long long tot = (long long)N * F;
    finalize_kernel<<<(unsigned)((tot + 255) / 256), 256, 0, stream>>>(out, b, tot, F,
                                                                       do_relu);
}

extern "C" void kernel_launch(void* const* d_in, const int* in_sizes, int n_in,
                              void* d_out, int out_size, void* d_ws, size_t ws_size,
                              hipStream_t stream) {
    const float* x      = (const float*)d_in[0];
    const float* W_rel  = (const float*)d_in[1];
    const float* b_rel  = (const float*)d_in[2];
    const float* W_root = (const float*)d_in[3];
    const float* W1  = (const float*)d_in[4];
    const float* as1 = (const float*)d_in[5];
    const float* ad1 = (const float*)d_in[6];
    const float* b1  = (const float*)d_in[7];
    const float* W2  = (const float*)d_in[8];
    const float* as2 = (const float*)d_in[9];
    const float* ad2 = (const float*)d_in[10];
    const float* b2  = (const float*)d_in[11];
    const float* W3  = (const float*)d_in[12];
    const float* as3 = (const float*)d_in[13];
    const float* ad3 = (const float*)d_in[14];
    const float* b3  = (const float*)d_in[15];
    const int*   ei  = (const int*)d_in[16];

    const int N = in_sizes[0] / 2;      // IN = 2
    const int E = in_sizes[16] / 2;     // edge_index is [2, E]
    const int HEADS = 4, HID = 64, OUT = 64;
    const int Fmax = HEADS * HID;       // 256

    // --- workspace carve-up (~120 MB) ---
    size_t big = (size_t)N * Fmax;                 // N*256 floats
    float*    P0    = (float*)d_ws;                // ping: layer input / out accum
    float*    P1    = P0 + big;                    // pong: GEMM result H
    float*    agg   = P1 + big;                    // N*2
    float*    al_s  = agg + (size_t)N * 2;         // N*heads (<= N*4)
    float*    al_d  = al_s + (size_t)N * 4;
    unsigned* m_enc = (unsigned*)(al_d + (size_t)N * 4);
    float*    den   = (float*)m_enc + (size_t)N * 4;
    float*    wbuf  = den + (size_t)N * 4;         // (E+N)*heads

    // ---- GraphConv(2 -> 64) + relu -> P0[:, :64] ----
    hipMemsetAsync(agg, 0, (size_t)N * 2 * sizeof(float), stream);
    gc_edge_kernel<<<(E + 255) / 256, 256, 0, stream>>>(x, ei, agg, E);
    gc_node_kernel<<<(N * 64 + 255) / 256, 256, 0, stream>>>(x, agg, W_rel, b_rel,
                                                             W_root, P0, N);

    // ---- GAT1: 64 -> 4x64 concat, relu ----   (in P0[N,64] -> out P0[N,256])
    run_gat_layer(P0, W1, as1, ad1, b1, N, E, /*K=*/HID, HEADS, HID, /*relu=*/1,
                  ei, P1, P0, al_s, al_d, m_enc, den, wbuf, stream);

    // ---- GAT2: 256 -> 4x64 concat, relu ----  (in P0 -> out P0)
    run_gat_layer(P0, W2, as2, ad2, b2, N, E, /*K=*/Fmax, HEADS, HID, /*relu=*/1,
                  ei, P1, P0, al_s, al_d, m_enc, den, wbuf, stream);

    // ---- GAT3: 256 -> 64, heads=1 (mean over 1 head == identity), no relu ----
    run_gat_layer(P0, W3, as3, ad3, b3, N, E, /*K=*/Fmax, /*heads=*/1, OUT,
                  /*relu=*/0, ei, P1, (float*)d_out, al_s, al_d, m_enc, den, wbuf,
                  stream);
}